// wvfn_71425306132798
// MI455X (gfx1250) — compile-verified
//
#include <hip/hip_runtime.h>
#include <math.h>

// ---------------------------------------------------------------------------
// VMC local-energy estimator for 2-particle hydrogenic wavefunction.
// Memory-bound streaming reduction: 48 MB in, 7 f32 sums out (~2.1 us floor
// at 23.3 TB/s on MI455X). Two walkers per thread -> 48B records ->
// 3x global_load_b128 (16B aligned). Wave-level reduction done on the matrix
// unit via V_WMMA_F32_16X16X4_F32 (f32 end-to-end, no precision loss).
// ---------------------------------------------------------------------------

typedef float v2f __attribute__((ext_vector_type(2)));
typedef float v8f __attribute__((ext_vector_type(8)));

#define NACC 7          // S11 S12 S21 S22 P11 P12 P22
#define Y00SQ 0.07957747154594767f   // 1/(4*pi)

// Reduce two per-lane f32 accumulators across the full wave32 using one
// 16x16x4 f32 WMMA per accumulator.
//   A (16x4 f32, 2 VGPRs):  VGPR0 = {K0: lanes0-15, K2: lanes16-31}
//                           VGPR1 = {K1: lanes0-15, K3: lanes16-31}
//   B = {1,0} selects K in {0,2} (acc0);  B = {0,1} selects K in {1,3} (acc1)
//   D[m,n] = acc[m] + acc[m+16]  ->  sum 8 D VGPRs + shfl_xor(16) = wave sum.
__device__ __forceinline__ void wave_reduce2(float a0, float a1,
                                             float& r0, float& r1) {
#if __has_builtin(__builtin_amdgcn_wmma_f32_16x16x4_f32)
    v2f A;  A[0] = a0;   A[1] = a1;
    v2f B0; B0[0] = 1.0f; B0[1] = 0.0f;
    v2f B1; B1[0] = 0.0f; B1[1] = 1.0f;
    v8f C = {};
    v8f D0 = __builtin_amdgcn_wmma_f32_16x16x4_f32(
        false, A, false, B0, (short)0, C, false, false);
    v8f D1 = __builtin_amdgcn_wmma_f32_16x16x4_f32(
        false, A, false, B1, (short)0, C, false, false);
    float t0 = ((D0[0]+D0[1])+(D0[2]+D0[3]))+((D0[4]+D0[5])+(D0[6]+D0[7]));
    float t1 = ((D1[0]+D1[1])+(D1[2]+D1[3]))+((D1[4]+D1[5])+(D1[6]+D1[7]));
#else
    float t0 = a0, t1 = a1;
    for (int off = 1; off < 16; off <<= 1) {
        t0 += __shfl_xor(t0, off, 32);
        t1 += __shfl_xor(t1, off, 32);
    }
#endif
    t0 += __shfl_xor(t0, 16, 32);
    t1 += __shfl_xor(t1, 16, 32);
    r0 = t0; r1 = t1;
}

__global__ void vmc_zero(float* partials) {
    if (threadIdx.x < 8) partials[threadIdx.x] = 0.0f;
}

struct Acc {
    float s11, s12, s21, s22, p11, p12, p22;
};

__device__ __forceinline__ void walker_accum(
        float x1, float y1, float z1, float x2, float y2, float z2,
        float m, float Z, float Z2, float K1, float K2, Acc& a) {
    const float rr1 = fmaf(x1, x1, fmaf(y1, y1, z1 * z1));
    const float rr2 = fmaf(x2, x2, fmaf(y2, y2, z2 * z2));
    const float ir1 = rsqrtf(rr1);
    const float ir2 = rsqrtf(rr2);
    const float r1  = rr1 * ir1;
    const float r2  = rr2 * ir2;

    const float dx = x1 - x2, dy = y1 - y2, dz = z1 - z2;
    const float dd = fmaf(dx, dx, fmaf(dy, dy, dz * dz));
    const float V  = -rsqrtf(dd);                    // -VB/d12, VB=1

    const float Eh = __expf(-0.5f * Z * (r1 + r2));  // exp(-Z*s/2)
    const float E  = Eh * Eh;                        // exp(-Z*s)

    const float Psi1 = (K1 * E) * m;
    const float Psi2 = (K2 * (2.0f - Z * r1) * (2.0f - Z * r2) * Eh) * m;

    // H_a = (-0.5*lap_a + V)*Psi_a ; lap1 = 2Z^2 - 2Z*sum(1/r),
    // lap2 = 0.5Z^2 - 2Z*sum(1/r)
    const float g  = V + Z * (ir1 + ir2);
    const float H1 = (g - Z2) * Psi1;
    const float H2 = (g - 0.25f * Z2) * Psi2;

    a.s11 = fmaf(Psi1, H1, a.s11);
    a.s12 = fmaf(Psi1, H2, a.s12);
    a.s21 = fmaf(Psi2, H1, a.s21);
    a.s22 = fmaf(Psi2, H2, a.s22);
    a.p11 = fmaf(Psi1, Psi1, a.p11);
    a.p12 = fmaf(Psi1, Psi2, a.p12);
    a.p22 = fmaf(Psi2, Psi2, a.p22);
}

__global__ void __launch_bounds__(256)
vmc_main(const float* __restrict__ Rs, const float* __restrict__ a_p,
         float* __restrict__ partials, int n, int iters) {
    const float Z  = 1.0f / a_p[0];
    const float Z2 = Z * Z;
    const float Z3 = Z2 * Z;
    const float K1 = 4.0f * Z3 * Y00SQ;      // Psi1 prefactor
    const float K2 = 0.125f * Z3 * Y00SQ;    // Psi2 prefactor

    const int tid = blockIdx.x * blockDim.x + threadIdx.x;
    const int T   = gridDim.x * blockDim.x;
    const int npairs = (n + 1) >> 1;         // 2 walkers per thread-iteration

    Acc acc = {0.f, 0.f, 0.f, 0.f, 0.f, 0.f, 0.f};

    const float4* __restrict__ R4 = (const float4*)Rs;

    // Uniform trip count for every thread: tail handled branchlessly by 0/1
    // masks so EXEC stays all-ones into the WMMA reduction.
    for (int k = 0; k < iters; ++k) {
        const int  i   = tid + k * T;               // pair index
        const int  idx = (i < npairs) ? i : 0;
        const float m0 = (2 * i     < n) ? 1.0f : 0.0f;
        const float m1 = (2 * i + 1 < n) ? 1.0f : 0.0f;

        // 48B per pair, 16B aligned -> 3x global_load_b128
        const float4* p = R4 + 3 * (size_t)idx;
        const float4 u0 = p[0];    // w0: x1 y1 z1 x2
        const float4 u1 = p[1];    // w0: y2 z2 | w1: x1 y1
        const float4 u2 = p[2];    // w1: z1 x2 y2 z2

        walker_accum(u0.x, u0.y, u0.z, u0.w, u1.x, u1.y, m0, Z, Z2, K1, K2, acc);
        walker_accum(u1.z, u1.w, u2.x, u2.y, u2.z, u2.w, m1, Z, Z2, K1, K2, acc);
    }

    // Wave32 reduction on the matrix unit (EXEC all ones here).
    float w[8];
    wave_reduce2(acc.s11, acc.s12, w[0], w[1]);
    wave_reduce2(acc.s21, acc.s22, w[2], w[3]);
    wave_reduce2(acc.p11, acc.p12, w[4], w[5]);
    wave_reduce2(acc.p22, 0.f,     w[6], w[7]);

    __shared__ float red[8][8];                 // [wave][acc], padded
    const int wave = threadIdx.x >> 5;
    const int lane = threadIdx.x & 31;
    if (lane == 0) {
        #pragma unroll
        for (int j = 0; j < NACC; ++j) red[wave][j] = w[j];
    }
    __syncthreads();

    if (threadIdx.x == 0) {
        const int nw = blockDim.x >> 5;
        float blk[NACC];
        #pragma unroll
        for (int j = 0; j < NACC; ++j) blk[j] = 0.0f;
        for (int wv = 0; wv < nw; ++wv)
            #pragma unroll
            for (int j = 0; j < NACC; ++j) blk[j] += red[wv][j];
        #pragma unroll
        for (int j = 0; j < NACC; ++j) atomicAdd(&partials[j], blk[j]);
    }
}

__global__ void vmc_final(const float* __restrict__ partials,
                          const float* c1re_p, const float* c1im_p,
                          const float* c2re_p, const float* c2im_p,
                          float* __restrict__ out, int out_size) {
    const float S11 = partials[0], S12 = partials[1];
    const float S21 = partials[2], S22 = partials[3];
    const float P11 = partials[4], P12 = partials[5], P22 = partials[6];

    const float c1r = c1re_p[0], c1i = c1im_p[0];
    const float c2r = c2re_p[0], c2i = c2im_p[0];

    const float A1 = c1r * c1r + c1i * c1i;          // |c1|^2
    const float A2 = c2r * c2r + c2i * c2i;          // |c2|^2
    const float u  = c1r * c2r + c1i * c2i;          // Re(conj(c1)*c2)
    const float v  = c1r * c2i - c1i * c2r;          // Im(conj(c1)*c2)

    // mean(hammy) = A1*S11 + (u+iv)*S12 + (u-iv)*S21 + A2*S22  (1/N cancels)
    const float Hre = A1 * S11 + u * (S12 + S21) + A2 * S22;
    const float Him = v * (S12 - S21);
    // mean(psi2)  = A1*P11 + 2u*P12 + A2*P22   (purely real)
    const float Dd  = A1 * P11 + 2.0f * u * P12 + A2 * P22;

    out[0] = Hre / Dd;
    if (out_size > 1) out[1] = Him / Dd;
}

extern "C" void kernel_launch(void* const* d_in, const int* in_sizes, int n_in,
                              void* d_out, int out_size, void* d_ws,
                              size_t ws_size, hipStream_t stream) {
    (void)n_in; (void)ws_size;
    const float* Rs   = (const float*)d_in[0];
    const float* a    = (const float*)d_in[1];
    const float* c1re = (const float*)d_in[2];
    const float* c1im = (const float*)d_in[3];
    const float* c2re = (const float*)d_in[4];
    const float* c2im = (const float*)d_in[5];
    float* partials   = (float*)d_ws;

    const int n      = in_sizes[0] / 6;     // walkers
    const int npairs = (n + 1) / 2;         // 2 walkers per thread-iteration
    const int threads = 256;                // 8 wave32 per block
    const int blocks  = 2048;
    const long long T = (long long)threads * blocks;
    int iters = (int)((npairs + T - 1) / T);
    if (iters < 1) iters = 1;

    vmc_zero<<<1, 32, 0, stream>>>(partials);
    vmc_main<<<blocks, threads, 0, stream>>>(Rs, a, partials, n, iters);
    vmc_final<<<1, 1, 0, stream>>>(partials, c1re, c1im, c2re, c2im,
                                   (float*)d_out, out_size);
}